// ComplexPadConv3Dt_9543417332124
// MI455X (gfx1250) — compile-verified
//
#include <hip/hip_runtime.h>

typedef __attribute__((ext_vector_type(2))) float v2f;
typedef __attribute__((ext_vector_type(8))) float v8f;

#define B_  2
#define T_  8
#define Z_  20
#define Y_  64
#define X_  64
#define C_  2
#define F1_ 32
#define F_  32
#define ZYX_ (Z_*Y_*X_)          /* 81920 */
#define MTOT_ (B_*T_*ZYX_)       /* 1310720 */
#define K1_ 108                  /* 27 taps * C=2 * (real,imag) */
#define K2_ 192                  /* 3 taps * F1=32 * (real,imag) */

/* workspace layout (float offsets) */
#define WS_B1   0                /* 54 pairs * 64 cols * 2 = 6912 floats  */
#define WS_B2   6912             /* 96 pairs * 64 cols * 2 = 12288 floats */
#define WS_INT  19200            /* MTOT_*64 floats intermediate          */

// ---------------------------------------------------------------------------
// Kernel 0: weight projection + pack GEMM B matrices (pair-swizzled for WMMA).
// B1[k][n] (108x64):  n<32 -> yr coeffs, n>=32 -> yi coeffs
//   k<54: [wsr | wsi],  k>=54: [-wsi | wsr]      (A = [xr patch ; xi patch])
// B2[k][n] (192x64):  k = dt*64 + part*32 + f1, part 0=spr 1=spi
//   part0: [wtr | wti], part1: [-wti | wtr]
// Storage: ws[(k>>1)*128 + n*2 + (k&1)] so kernels read float2 = (B[k],B[k+1]).
// ---------------------------------------------------------------------------
__global__ void __launch_bounds__(64)
proj_pack_kernel(const float* __restrict__ wxyz_r, const float* __restrict__ wxyz_i,
                 const float* __restrict__ wt_r,   const float* __restrict__ wt_i,
                 float* __restrict__ ws) {
  int tid = threadIdx.x;
  float* B1 = ws + WS_B1;
  float* B2 = ws + WS_B2;
  if (tid < 32) {
    int f1 = tid;
    // spatial filter f1: zero-mean + bounded-norm over 54 complex taps
    float sr = 0.f, si = 0.f;
    for (int k = 0; k < 54; ++k) {
      int idx = (k >> 1) * (C_ * F1_) + (k & 1) * F1_ + f1;   // (spatial, c, f1)
      sr += wxyz_r[idx]; si += wxyz_i[idx];
    }
    float mr = sr * (1.f / 54.f), mi = si * (1.f / 54.f);
    float n2 = 0.f;
    for (int k = 0; k < 54; ++k) {
      int idx = (k >> 1) * (C_ * F1_) + (k & 1) * F1_ + f1;
      float a = wxyz_r[idx] - mr, b = wxyz_i[idx] - mi;
      n2 += a * a + b * b;
    }
    float s = 1.f / fmaxf(sqrtf(n2), 1.f);
    for (int k = 0; k < K1_; ++k) {
      int kk = (k < 54) ? k : (k - 54);
      int idx = (kk >> 1) * (C_ * F1_) + (kk & 1) * F1_ + f1;
      float ar = (wxyz_r[idx] - mr) * s;
      float ai = (wxyz_i[idx] - mi) * s;
      float vr = (k < 54) ? ar : -ai;   // coeff toward yr
      float vi = (k < 54) ? ai :  ar;   // coeff toward yi
      int base = (k >> 1) * 128 + (k & 1);
      B1[base + f1 * 2]        = vr;
      B1[base + (32 + f1) * 2] = vi;
    }
  } else {
    int f = tid - 32;
    // temporal filter f: bounded-norm (no zero-mean) over 3*32 complex taps
    float n2 = 0.f;
    for (int dt = 0; dt < 3; ++dt)
      for (int f1 = 0; f1 < F1_; ++f1) {
        float a = wt_r[dt * F1_ * F_ + f1 * F_ + f];
        float b = wt_i[dt * F1_ * F_ + f1 * F_ + f];
        n2 += a * a + b * b;
      }
    float s = 1.f / fmaxf(sqrtf(n2), 1.f);
    for (int k = 0; k < K2_; ++k) {
      int dt = k >> 6, part = (k >> 5) & 1, f1 = k & 31;
      float ar = wt_r[dt * F1_ * F_ + f1 * F_ + f] * s;
      float ai = wt_i[dt * F1_ * F_ + f1 * F_ + f] * s;
      float vr = part ? -ai : ar;
      float vi = part ?  ar : ai;
      int base = (k >> 1) * 128 + (k & 1);
      B2[base + f * 2]        = vr;
      B2[base + (32 + f) * 2] = vi;
    }
  }
}

// ---------------------------------------------------------------------------
// Kernel 1: spatial complex 3x3x3 conv as implicit GEMM, f32 WMMA 16x16x4.
// One wave = one 16-wide x-tile; produces 16x64 (spr||spi) into intermediate.
// All tap address components are hoisted: per k-step addressing is just
// cndmask-selects between precomputed registers + one add.
// ---------------------------------------------------------------------------
__global__ void __launch_bounds__(256)
spatial_wmma_kernel(const float* __restrict__ xr, const float* __restrict__ xi,
                    const float* __restrict__ ws, float* __restrict__ inter) {
  __shared__ __align__(16) float ldsB[54 * 128];          // 27,648 B
  for (int i = threadIdx.x; i < 54 * 128; i += 256) ldsB[i] = ws[WS_B1 + i];
  __syncthreads();
  const v2f* Bp = (const v2f*)ldsB;

  const int lane = threadIdx.x & 31;
  const int wave = threadIdx.x >> 5;
  const int hi = lane >> 4;      // half-wave: selects K sub-pair
  const int n  = lane & 15;      // N column within tile / A row m for this lane
  const int mt = blockIdx.x * 8 + wave;
  const int p0 = mt * 16;        // 16 consecutive x positions (row-aligned)

  const int x0 = p0 & 63;
  const int rest = p0 >> 6;      // (bt*20+z)*64 + y
  const int y  = rest & 63;
  const int rest2 = rest >> 6;   // bt*20 + z
  const int z  = rest2 % 20;
  const int bt = rest2 / 20;     // b*8 + t
  const int x  = x0 + n;

  // hoisted clamped (== symmetric, pad 1) address components, float units
  unsigned rowoff[9];
#pragma unroll
  for (int dz = 0; dz < 3; ++dz)
#pragma unroll
    for (int dy = 0; dy < 3; ++dy) {
      int zz = min(max(z + dz - 1, 0), Z_ - 1);
      int yy = min(max(y + dy - 1, 0), Y_ - 1);
      rowoff[dz * 3 + dy] = (unsigned)(((bt * Z_ + zz) * Y_ + yy) * X_) * C_;
    }
  unsigned xoff[3];
  xoff[0] = (unsigned)max(x - 1, 0) * C_;
  xoff[1] = (unsigned)x * C_;
  xoff[2] = (unsigned)min(x + 1, X_ - 1) * C_;

  v8f acc0 = {}, acc1 = {}, acc2 = {}, acc3 = {};

#pragma unroll
  for (int ks = 0; ks < 27; ++ks) {           // K = 108 = 27 * 4
    const int s0 = 2 * ks, s1 = 2 * ks + 1;   // pair indices for hi=0 / hi=1
    const int o0 = (s0 < 27) ? s0 : s0 - 27;  // compile-time tap indices
    const int o1 = (s1 < 27) ? s1 : s1 - 27;
    unsigned off = hi ? (rowoff[o1 / 3] + xoff[o1 % 3])
                      : (rowoff[o0 / 3] + xoff[o0 % 3]);
    const float* src;
    if (s1 < 27)       src = xr;              // both halves real
    else if (s0 >= 27) src = xi;              // both halves imag
    else               src = hi ? xi : xr;    // boundary k-step only
    v2f a = *(const v2f*)(src + off);         // (c=0, c=1) contiguous

    const int s = 2 * ks + 0;                 // LDS pair row base for hi=0
    const v2f b0 = Bp[(s + hi) * 64 +  0 + n];
    const v2f b1 = Bp[(s + hi) * 64 + 16 + n];
    const v2f b2 = Bp[(s + hi) * 64 + 32 + n];
    const v2f b3 = Bp[(s + hi) * 64 + 48 + n];
    acc0 = __builtin_amdgcn_wmma_f32_16x16x4_f32(false, a, false, b0, (short)0, acc0, false, false);
    acc1 = __builtin_amdgcn_wmma_f32_16x16x4_f32(false, a, false, b1, (short)0, acc1, false, false);
    acc2 = __builtin_amdgcn_wmma_f32_16x16x4_f32(false, a, false, b2, (short)0, acc2, false, false);
    acc3 = __builtin_amdgcn_wmma_f32_16x16x4_f32(false, a, false, b3, (short)0, acc3, false, false);
  }

#pragma unroll
  for (int r = 0; r < 8; ++r) {
    unsigned po = (unsigned)(p0 + r + hi * 8);  // C/D layout: lanes16-31 = M+8
    inter[po * 64 +  0 + n] = acc0[r];
    inter[po * 64 + 16 + n] = acc1[r];
    inter[po * 64 + 32 + n] = acc2[r];
    inter[po * 64 + 48 + n] = acc3[r];
  }
}

// ---------------------------------------------------------------------------
// Kernel 2: temporal complex 3x1x1 conv as GEMM (K=192), f32 WMMA 16x16x4.
// Reads intermediate at t-1/t/t+1 (symmetric pad on T), writes yr then yi.
// ---------------------------------------------------------------------------
__global__ void __launch_bounds__(256)
temporal_wmma_kernel(const float* __restrict__ inter, const float* __restrict__ ws,
                     float* __restrict__ out) {
  __shared__ __align__(16) float ldsB[96 * 128];          // 49,152 B
  for (int i = threadIdx.x; i < 96 * 128; i += 256) ldsB[i] = ws[WS_B2 + i];
  __syncthreads();
  const v2f* Bp = (const v2f*)ldsB;

  const int lane = threadIdx.x & 31;
  const int wave = threadIdx.x >> 5;
  const int hi = lane >> 4;
  const int n  = lane & 15;
  const int mt = blockIdx.x * 8 + wave;
  const int p0 = mt * 16;
  const int p  = p0 + n;                    // this lane's A row position
  const int t  = (p0 / ZYX_) & 7;           // uniform within tile

  // hoisted per-dt row bases of the intermediate, in float units
  unsigned rbase[3];
  rbase[0] = (unsigned)(p + ((t == 0)      ? 0 : -ZYX_)) * 64u;  // symmetric -1 -> 0
  rbase[1] = (unsigned)p * 64u;
  rbase[2] = (unsigned)(p + ((t == T_ - 1) ? 0 :  ZYX_)) * 64u;  // symmetric T -> T-1

  v8f acc0 = {}, acc1 = {}, acc2 = {}, acc3 = {};

#pragma unroll
  for (int ks = 0; ks < 48; ++ks) {         // K = 192 = 48 * 4
    const int kb0 = 4 * ks, kb1 = 4 * ks + 2;
    const int dt0 = kb0 >> 6, dt1 = kb1 >> 6;   // compile-time 0..2
    const int c0  = kb0 & 63, c1  = kb1 & 63;   // even, never cross 64 boundary
    unsigned off = hi ? (rbase[dt1] + c1) : (rbase[dt0] + c0);
    v2f a = *(const v2f*)(inter + off);

    const int s = kb0 >> 1;                 // LDS pair row base for hi=0
    const v2f b0 = Bp[(s + hi) * 64 +  0 + n];
    const v2f b1 = Bp[(s + hi) * 64 + 16 + n];
    const v2f b2 = Bp[(s + hi) * 64 + 32 + n];
    const v2f b3 = Bp[(s + hi) * 64 + 48 + n];
    acc0 = __builtin_amdgcn_wmma_f32_16x16x4_f32(false, a, false, b0, (short)0, acc0, false, false);
    acc1 = __builtin_amdgcn_wmma_f32_16x16x4_f32(false, a, false, b1, (short)0, acc1, false, false);
    acc2 = __builtin_amdgcn_wmma_f32_16x16x4_f32(false, a, false, b2, (short)0, acc2, false, false);
    acc3 = __builtin_amdgcn_wmma_f32_16x16x4_f32(false, a, false, b3, (short)0, acc3, false, false);
  }

  float* yr = out;
  float* yi = out + (long)MTOT_ * F_;
#pragma unroll
  for (int r = 0; r < 8; ++r) {
    unsigned po = (unsigned)(p0 + r + hi * 8);
    yr[po * 32 +      n] = acc0[r];
    yr[po * 32 + 16 + n] = acc1[r];
    yi[po * 32 +      n] = acc2[r];
    yi[po * 32 + 16 + n] = acc3[r];
  }
}

// ---------------------------------------------------------------------------
extern "C" void kernel_launch(void* const* d_in, const int* in_sizes, int n_in,
                              void* d_out, int out_size, void* d_ws, size_t ws_size,
                              hipStream_t stream) {
  const float* xr     = (const float*)d_in[0];
  const float* xi     = (const float*)d_in[1];
  const float* wxyz_r = (const float*)d_in[2];
  const float* wxyz_i = (const float*)d_in[3];
  const float* wt_r   = (const float*)d_in[4];
  const float* wt_i   = (const float*)d_in[5];
  float* ws  = (float*)d_ws;                // needs (19200 + MTOT*64)*4 ≈ 336 MB
  float* out = (float*)d_out;

  proj_pack_kernel<<<1, 64, 0, stream>>>(wxyz_r, wxyz_i, wt_r, wt_i, ws);

  const int blocks = MTOT_ / (16 * 8);      // 10240, one wave per 16-pos tile
  spatial_wmma_kernel<<<blocks, 256, 0, stream>>>(xr, xi, ws, ws + WS_INT);
  temporal_wmma_kernel<<<blocks, 256, 0, stream>>>(ws + WS_INT, ws, out);
}